// GCNK_40956808135032
// MI455X (gfx1250) — compile-verified
//
#include <hip/hip_runtime.h>
#include <math.h>

#define NNODES 100000
#define NEDGES 1600000
#define NFEAT  512
#define NHID   256
#define NCLASS 40
#define NCPAD  48   // NCLASS padded to 3 WMMA tiles

typedef __attribute__((ext_vector_type(2))) float v2f;
typedef __attribute__((ext_vector_type(8))) float v8f;

// ---------------------------------------------------------------------------
// fp32 WMMA GEMM: C[M,*] = A[M,K] @ Bp[K,*] + bias
// Bp is "fragment-major" packed: element (k,col) at Bp[(k>>1)*2*ldb + col*2 + (k&1)]
// so each lane's B fragment (rows k,k+1 of one column) is one aligned b64 load.
// One wave computes a 16(M) x (NT*16)(N) tile with V_WMMA_F32_16X16X4_F32,
// software-pipelined: next K-step's A/B fragments load while wmmas issue.
// M multiple of 16 (100000 = 6250*16); 8 waves (8 M-tiles) per 256-thread block.
// ---------------------------------------------------------------------------
template<int NT>
__global__ __launch_bounds__(256)
void gemm_wmma_f32(const float* __restrict__ A, const float* __restrict__ Bp,
                   const float* __restrict__ bias, float* __restrict__ C,
                   int M, int K, int ldb, int ldc) {
  const int lane   = threadIdx.x & 31;
  const int wave   = threadIdx.x >> 5;
  const int mtiles = M >> 4;
  const int tile_m = blockIdx.x * 8 + wave;
  if (tile_m >= mtiles) return;              // wave-uniform exit: EXEC stays full

  const int row0  = tile_m << 4;
  const int col0  = blockIdx.y * (NT * 16);
  const int mrow  = lane & 15;               // A: M index for this lane
  const int khalf = (lane >> 4) << 1;        // K offset 0 or 2 per half-wave

  v8f acc[NT];
#pragma unroll
  for (int t = 0; t < NT; ++t) {
    float bv = bias[col0 + t * 16 + (lane & 15)];
    v8f ini = {bv, bv, bv, bv, bv, bv, bv, bv};
    acc[t] = ini;
  }

  const float* arow  = A  + (size_t)(row0 + mrow) * K + khalf;
  const float* bbase = Bp + (size_t)khalf * ldb + (size_t)(col0 + (lane & 15)) * 2;

  // Prologue: fragments for k0 = 0
  v2f a = *(const v2f*)(arow);
  v2f b[NT];
#pragma unroll
  for (int t = 0; t < NT; ++t)
    b[t] = *(const v2f*)(bbase + t * 32);

  // Pipelined main loop: load k0+4 while computing k0
#pragma unroll 2
  for (int k0 = 0; k0 < K - 4; k0 += 4) {
    const int kn = k0 + 4;
    v2f a_n = *(const v2f*)(arow + kn);
    v2f b_n[NT];
#pragma unroll
    for (int t = 0; t < NT; ++t)
      b_n[t] = *(const v2f*)(bbase + (size_t)kn * ldb + t * 32);
#pragma unroll
    for (int t = 0; t < NT; ++t)
      acc[t] = __builtin_amdgcn_wmma_f32_16x16x4_f32(
          false, a, false, b[t], (short)0, acc[t], false, false);
    a = a_n;
#pragma unroll
    for (int t = 0; t < NT; ++t) b[t] = b_n[t];
  }
  // Epilogue: last K-step
#pragma unroll
  for (int t = 0; t < NT; ++t)
    acc[t] = __builtin_amdgcn_wmma_f32_16x16x4_f32(
        false, a, false, b[t], (short)0, acc[t], false, false);

  // C/D layout: VGPR v = rows (v, v+8); lane halves select row groups
  const int rbase = row0 + ((lane >> 4) << 3);
#pragma unroll
  for (int t = 0; t < NT; ++t) {
    int col = col0 + t * 16 + (lane & 15);
    float* cp = C + (size_t)rbase * ldc + col;
#pragma unroll
    for (int v = 0; v < 8; ++v)
      cp[(size_t)v * ldc] = acc[t][v];
  }
}

// Pack B [K,N] row-major -> fragment-major Bp (pair of K-rows interleaved).
__global__ void pack_b(const float* __restrict__ B, float* __restrict__ Bp,
                       int K, int N) {
  int i = blockIdx.x * blockDim.x + threadIdx.x;
  if (i >= K * N) return;
  int k = i / N, col = i - k * N;
  Bp[(size_t)(k >> 1) * (2 * N) + col * 2 + (k & 1)] = B[i];
}

// Pad W2 [256,40] -> 48 cols and pack fragment-major; pad b2 -> b2p[48].
__global__ void pad_pack_w2(const float* __restrict__ W2, const float* __restrict__ b2,
                            float* __restrict__ W2pp, float* __restrict__ b2p) {
  int i = blockIdx.x * blockDim.x + threadIdx.x;
  if (i < NHID * NCPAD) {
    int k = i / NCPAD, c = i - k * NCPAD;
    float v = (c < NCLASS) ? W2[k * NCLASS + c] : 0.0f;
    W2pp[(size_t)(k >> 1) * (2 * NCPAD) + c * 2 + (k & 1)] = v;
  }
  if (i < NCPAD) b2p[i] = (i < NCLASS) ? b2[i] : 0.0f;
}

// CSR row pointers from sorted tgt: row_ptr[n] = lower_bound(tgt, n)
__global__ void build_rowptr(const int* __restrict__ tgt, int E, int Nn,
                             int* __restrict__ row_ptr) {
  int n = blockIdx.x * blockDim.x + threadIdx.x;
  if (n > Nn) return;
  int lo = 0, hi = E;
  while (lo < hi) {
    int mid = (lo + hi) >> 1;
    if (tgt[mid] < n) lo = mid + 1; else hi = mid;
  }
  row_ptr[n] = lo;
}

// Layer-1 aggregation + ReLU: one block (256 threads = NHID feats) per node,
// register accumulation over in-edges -> no atomics. h_lin is L2-resident.
__global__ __launch_bounds__(NHID)
void agg_relu(const float* __restrict__ hlin, const int* __restrict__ src,
              const float* __restrict__ mvals, const int* __restrict__ row_ptr,
              float* __restrict__ out) {
  const int n = blockIdx.x;
  const int f = threadIdx.x;
  const int e0 = row_ptr[n], e1 = row_ptr[n + 1];
  float acc = 0.0f;
  for (int e = e0; e < e1; ++e) {
    if (e + 1 < e1)  // gfx1250 global_prefetch_b8 for next gathered row
      __builtin_prefetch(&hlin[(size_t)src[e + 1] * NHID + f], 0, 3);
    acc = fmaf(hlin[(size_t)src[e] * NHID + f], mvals[e], acc);
  }
  out[(size_t)n * NHID + f] = fmaxf(acc, 0.0f);
}

// Layer-2 aggregation fused with log_softmax: 64 threads / node, 40 classes,
// reading the 48-column padded lin2.
__global__ __launch_bounds__(64)
void agg_logsoftmax(const float* __restrict__ lin2p, const int* __restrict__ src,
                    const float* __restrict__ mvals, const int* __restrict__ row_ptr,
                    float* __restrict__ out) {
  __shared__ float sm[NCLASS];
  __shared__ float s_max, s_lse;
  const int n = blockIdx.x;
  const int c = threadIdx.x;
  const int e0 = row_ptr[n], e1 = row_ptr[n + 1];
  float acc = 0.0f;
  if (c < NCLASS) {
    for (int e = e0; e < e1; ++e)
      acc = fmaf(lin2p[(size_t)src[e] * NCPAD + c], mvals[e], acc);
    sm[c] = acc;
  }
  __syncthreads();
  if (threadIdx.x == 0) {
    float m = -INFINITY;
    for (int i = 0; i < NCLASS; ++i) m = fmaxf(m, sm[i]);
    float s = 0.0f;
    for (int i = 0; i < NCLASS; ++i) s += expf(sm[i] - m);
    s_max = m;
    s_lse = logf(s);
  }
  __syncthreads();
  if (c < NCLASS)
    out[(size_t)n * NCLASS + c] = acc - s_max - s_lse;
}

extern "C" void kernel_launch(void* const* d_in, const int* in_sizes, int n_in,
                              void* d_out, int out_size, void* d_ws, size_t ws_size,
                              hipStream_t stream) {
  const float* x     = (const float*)d_in[0];   // [NNODES, NFEAT]
  const int*   src   = (const int*)  d_in[1];   // [NEDGES]
  const int*   tgt   = (const int*)  d_in[2];   // [NEDGES] sorted
  const float* mvals = (const float*)d_in[3];   // [NEDGES]
  const float* W1    = (const float*)d_in[4];   // [NFEAT, NHID]
  const float* b1    = (const float*)d_in[5];   // [NHID]
  const float* W2    = (const float*)d_in[6];   // [NHID, NCLASS]
  const float* b2    = (const float*)d_in[7];   // [NCLASS]
  float* out = (float*)d_out;                    // [NNODES, NCLASS]

  // Workspace carve-up (offsets stay 256B-aligned):
  char* ws = (char*)d_ws;
  size_t off = 0;
  float* hlin  = (float*)(ws + off); off += (size_t)NNODES * NHID  * 4;  // 102.4 MB
  float* h1    = (float*)(ws + off); off += (size_t)NNODES * NHID  * 4;  // 102.4 MB
  float* lin2p = (float*)(ws + off); off += (size_t)NNODES * NCPAD * 4;  //  19.2 MB
  int*   row_ptr = (int*)(ws + off); off += (size_t)(NNODES + 1) * 4;
  off = (off + 255) & ~(size_t)255;
  float* W1p   = (float*)(ws + off); off += (size_t)NFEAT * NHID * 4;    // 512 KB
  float* W2pp  = (float*)(ws + off); off += (size_t)NHID * NCPAD * 4;    //  49 KB
  float* b2p   = (float*)(ws + off); off += (size_t)NCPAD * 4;

  // Independent prep kernels first
  build_rowptr<<<(NNODES + 1 + 255) / 256, 256, 0, stream>>>(
      tgt, NEDGES, NNODES, row_ptr);
  pack_b<<<(NFEAT * NHID + 255) / 256, 256, 0, stream>>>(W1, W1p, NFEAT, NHID);
  pad_pack_w2<<<(NHID * NCPAD + 255) / 256, 256, 0, stream>>>(W2, b2, W2pp, b2p);

  // GEMM1: [100000,512] @ [512,256] + b1 -> hlin
  {
    int mtiles = NNODES / 16;                        // 6250
    dim3 grid((mtiles + 7) / 8, NHID / (8 * 16));    // (782, 2)
    gemm_wmma_f32<8><<<grid, 256, 0, stream>>>(
        x, W1p, b1, hlin, NNODES, NFEAT, NHID, NHID);
  }

  // Aggregation 1 + ReLU -> h1
  agg_relu<<<NNODES, NHID, 0, stream>>>(hlin, src, mvals, row_ptr, h1);

  // GEMM2: [100000,256] @ [256,48(pad)] + b2p -> lin2p
  {
    int mtiles = NNODES / 16;
    dim3 grid((mtiles + 7) / 8, 1);
    gemm_wmma_f32<3><<<grid, 256, 0, stream>>>(
        h1, W2pp, b2p, lin2p, NNODES, NHID, NCPAD, NCPAD);
  }

  // Aggregation 2 + log_softmax -> out
  agg_logsoftmax<<<NNODES, 64, 0, stream>>>(lin2p, src, mvals, row_ptr, out);
}